// SemiLoss_43757126811747
// MI455X (gfx1250) — compile-verified
//
#include <hip/hip_runtime.h>

typedef __attribute__((ext_vector_type(2))) float v2f;
typedef __attribute__((ext_vector_type(4))) float v4f;
typedef __attribute__((ext_vector_type(8))) float v8f;

#define L2CLAMP  -144.26950408889634f   // -100 / ln(2): log-clamp in log2 domain
#define LN2       0.6931471805599453f
#define INV_T     0.25f                 // 1/TEMPERATURE
#define THRESH    0.5f

#if __has_builtin(__builtin_amdgcn_logf)
  #define LOG2F(x) __builtin_amdgcn_logf(x)     // v_log_f32 (native log2, TRANS pipe)
#else
  #define LOG2F(x) __log2f(x)
#endif

// ---------------------------------------------------------------------------
// Wave32 cross-lane sum on the matrix pipe (gfx1250 WMMA).
// A(16x4 f32) = {acc, 0} per lane -> A[M][0]=acc[M], A[M][2]=acc[M+16], rest 0
// B(4x16 f32) = ones. D[m][n] = acc[m] + acc[m+16] for every column n.
// Lane L in [0,15] holds D[0..7][L]; lane L+16 holds D[8..15][L].
// Wave total = per-lane sum(c[0..7]) + SWAPX16 partner (single ds_swizzle).
// EXEC is all-ones at every call site (unconditional, full 256-thread blocks).
// ---------------------------------------------------------------------------
__device__ __forceinline__ float swap16(float x) {
    // group-of-32 swizzle: xor=0x10, or=0, and=0x1f -> offset 0x401F (SWAPX16)
    return __int_as_float(__builtin_amdgcn_ds_swizzle(__float_as_int(x), 0x401F));
}

__device__ __forceinline__ float wave_reduce_wmma(float acc) {
    v2f a; a.x = acc;  a.y = 0.0f;
    v2f b; b.x = 1.0f; b.y = 1.0f;
    v8f c = {};
    c = __builtin_amdgcn_wmma_f32_16x16x4_f32(
            /*neg_a=*/false, a, /*neg_b=*/false, b,
            /*c_mod=*/(short)0, c, /*reuse_a=*/false, /*reuse_b=*/false);
    float s = ((c[0] + c[1]) + (c[2] + c[3])) + ((c[4] + c[5]) + (c[6] + c[7]));
    return s + swap16(s);               // full 32-lane total in every lane
}

// Per-element loss, log2-domain BCE. mln2 = mask * ln(2) (0 or ln2, per row).
// bce = -ln2 * (lb*(p2-q2) + q2);  pseudo+consistency = d1^2 + d2^2.
__device__ __forceinline__ float elem_loss(float o, float oc, float lb, float mln2) {
    float om = 1.0f - o;
    float p2 = fmaxf(LOG2F(o),  L2CLAMP);       // clamped log2(o)
    float q2 = fmaxf(LOG2F(om), L2CLAMP);       // clamped log2(1-o)
    float w  = fmaf(lb, p2 - q2, q2);
    bool  hi = o > THRESH;
    float t  = hi ? om : o;                     // |o - sharp| * TEMPERATURE
    float d1 = t * INV_T;                       // (o - sharp), up to sign
    float sharp = hi ? fmaf(om, INV_T, o) : o * (1.0f - INV_T);
    float d2 = oc - sharp;
    return fmaf(mln2, -w, fmaf(d1, d1, d2 * d2));
}

// UNIFORM_ROW: row index is wave-uniform (host guarantees c4 is a power of two
// >= 32, so no wave of 32 consecutive float4 indices straddles a row). The
// readfirstlane broadcast turns src[] into an s_load and the mask into SALU.
template <bool UNIFORM_ROW>
__global__ void __launch_bounds__(256)
semiloss_partial_kernel(const float* __restrict__ out,
                        const float* __restrict__ outc,
                        const float* __restrict__ lab,
                        const int*   __restrict__ src,
                        float* __restrict__ partials,
                        int n4, int c4, int c4shift) {
    const int stride = gridDim.x * blockDim.x;
    float acc = 0.0f;
    // Streaming phase: 3x global_load_b128 NT per iteration. 402MB single-use
    // stream > 192MB L2, so non-temporal is the right cache policy.
    for (int i = blockIdx.x * blockDim.x + threadIdx.x; i < n4; i += stride) {
        v4f o  = __builtin_nontemporal_load((const v4f*)out  + i);
        v4f oc = __builtin_nontemporal_load((const v4f*)outc + i);
        v4f lb = __builtin_nontemporal_load((const v4f*)lab  + i);
        float mln2;
        if (UNIFORM_ROW) {
            int row = __builtin_amdgcn_readfirstlane(i) >> c4shift;  // scalar
            int s = src[row];                                        // s_load
            mln2 = (s == 0 || s == 1) ? LN2 : 0.0f;                  // SALU
        } else {
            int row = (unsigned)i / (unsigned)c4;
            int s = src[row];
            mln2 = (s == 0 || s == 1) ? LN2 : 0.0f;
        }
        acc += elem_loss(o.x, oc.x, lb.x, mln2);
        acc += elem_loss(o.y, oc.y, lb.y, mln2);
        acc += elem_loss(o.z, oc.z, lb.z, mln2);
        acc += elem_loss(o.w, oc.w, lb.w, mln2);
    }
    // Epilogue: WMMA cross-lane reduce, then LDS across the 8 waves.
    float wsum = wave_reduce_wmma(acc);
    __shared__ float lds[8];
    const int lane = threadIdx.x & 31;
    const int wid  = threadIdx.x >> 5;
    if (lane == 0) lds[wid] = wsum;
    __syncthreads();
    if (threadIdx.x == 0) {
        float b = 0.0f;
        #pragma unroll
        for (int w = 0; w < 8; ++w) b += lds[w];
        partials[blockIdx.x] = b;
    }
}

__global__ void __launch_bounds__(256)
semiloss_final_kernel(const float* __restrict__ partials, int n,
                      float* __restrict__ result, float invB) {
    float acc = 0.0f;
    for (int i = threadIdx.x; i < n; i += blockDim.x) acc += partials[i];
    float wsum = wave_reduce_wmma(acc);
    __shared__ float lds[8];
    const int lane = threadIdx.x & 31;
    const int wid  = threadIdx.x >> 5;
    if (lane == 0) lds[wid] = wsum;
    __syncthreads();
    if (threadIdx.x == 0) {
        float t = 0.0f;
        #pragma unroll
        for (int w = 0; w < 8; ++w) t += lds[w];
        result[0] = t * invB;
    }
}

extern "C" void kernel_launch(void* const* d_in, const int* in_sizes, int n_in,
                              void* d_out, int out_size, void* d_ws, size_t ws_size,
                              hipStream_t stream) {
    const float* out  = (const float*)d_in[0];
    const float* outc = (const float*)d_in[1];
    const float* lab  = (const float*)d_in[2];
    const int*   src  = (const int*)d_in[3];

    const int N  = in_sizes[0];          // B*C = 8192*4096
    const int B  = in_sizes[3];          // 8192
    const int C  = N / B;                // 4096
    const int n4 = N >> 2;
    const int c4 = C >> 2;

    // Wave-uniform row path requires c4 = power of two and a multiple of the
    // wave width (32), so a wave's 32 consecutive indices share one row.
    int c4shift = -1;
    if (c4 >= 32 && (c4 & (c4 - 1)) == 0) {
        c4shift = 0;
        while ((1 << c4shift) < c4) ++c4shift;
    }

    // 2048 block partials (8 KB) in workspace -> deterministic two-pass sum.
    int nblocks = 2048;
    const int maxp = (int)(ws_size / sizeof(float));
    if (maxp > 0 && nblocks > maxp) nblocks = maxp;
    float* partials = (float*)d_ws;

    if (c4shift >= 0) {
        semiloss_partial_kernel<true><<<nblocks, 256, 0, stream>>>(
            out, outc, lab, src, partials, n4, c4, c4shift);
    } else {
        semiloss_partial_kernel<false><<<nblocks, 256, 0, stream>>>(
            out, outc, lab, src, partials, n4, c4, c4shift);
    }
    semiloss_final_kernel<<<1, 256, 0, stream>>>(
        partials, nblocks, (float*)d_out, 1.0f / (float)B);
}